// ClusterNet_35940286333533
// MI455X (gfx1250) — compile-verified
//
#include <hip/hip_runtime.h>
#include <hip/hip_bf16.h>

#define NN 100000
#define NE 1600000
#define NG 64

typedef float v2f_t __attribute__((ext_vector_type(2)));
typedef float v8f_t __attribute__((ext_vector_type(8)));

// ---------------------------------------------------------------- degree count
__global__ void k_count(const int* __restrict__ dst, float* __restrict__ cnt) {
    int e = blockIdx.x * blockDim.x + threadIdx.x;
    if (e < NE) atomicAdd(&cnt[dst[e]], 1.0f);
}

__global__ void k_inv(float* __restrict__ cnt) {
    int n = blockIdx.x * blockDim.x + threadIdx.x;
    if (n < NN) {
        float c = cnt[n];
        cnt[n] = (c > 0.0f) ? (1.0f / c) : 0.0f;
    }
}

// --------------------------------------------------- edge scatter-add (to L2)
// one wave per edge; lane handles feature(s). acc must be pre-zeroed.
template <int F>
__global__ void k_aggregate(const float* __restrict__ h, const int* __restrict__ src,
                            const int* __restrict__ dst, float* __restrict__ acc) {
    int wave   = (blockIdx.x * blockDim.x + threadIdx.x) >> 5;
    int lane   = threadIdx.x & 31;
    int nwaves = (gridDim.x * blockDim.x) >> 5;
    for (int e = wave; e < NE; e += nwaves) {
        int s = src[e];
        int d = dst[e];
        const float* hs = h + (size_t)s * F;
        float*       ad = acc + (size_t)d * F;
        atomicAdd(&ad[lane], hs[lane]);
        if (F == 64) atomicAdd(&ad[lane + 32], hs[lane + 32]);
    }
}

// ------------------------------------------------- fused SAGE GEMM via WMMA f32
// hout[16xN block] = (acc * inv) @ Wl + bl + hin @ Wr
// one wave handles a 16-row node block, full N=64 output width.
template <int K>
__global__ void __launch_bounds__(256)
k_sage_gemm(const float* __restrict__ acc, const float* __restrict__ inv,
            const float* __restrict__ hin, const float* __restrict__ Wl,
            const float* __restrict__ bl,  const float* __restrict__ Wr,
            float* __restrict__ hout) {
    const int NB = NN / 16;  // 6250 blocks of 16 nodes
    int wave = (blockIdx.x * blockDim.x + threadIdx.x) >> 5;
    if (wave >= NB) return;                       // wave-uniform: EXEC stays all-1s
    int lane = threadIdx.x & 31;
    int m    = lane & 15;                         // row within tile (both lane halves)
    int kh   = lane >> 4;                         // K-half selector
    int row0 = wave * 16;
    int arow = row0 + m;
    float sc = inv[arow];                         // fold mean-division into A load
    const float* accR = acc + (size_t)arow * K;
    const float* hinR = hin + (size_t)arow * K;

    v8f_t cacc[4] = {v8f_t{}, v8f_t{}, v8f_t{}, v8f_t{}};
    for (int k = 0; k < K; k += 4) {
        int ka = k + 2 * kh;                      // A layout: lanes>=16 hold K+2,K+3
        v2f_t aM, aH;
        aM.x = accR[ka] * sc;
        aM.y = accR[ka + 1] * sc;
        aH.x = hinR[ka];
        aH.y = hinR[ka + 1];
#pragma unroll
        for (int nt = 0; nt < 4; ++nt) {
            int col = nt * 16 + m;
            v2f_t bL, bR;                         // B layout: vgpr j = row ka+j
            bL.x = Wl[ka * 64 + col];
            bL.y = Wl[(ka + 1) * 64 + col];
            bR.x = Wr[ka * 64 + col];
            bR.y = Wr[(ka + 1) * 64 + col];
            cacc[nt] = __builtin_amdgcn_wmma_f32_16x16x4_f32(
                false, aM, false, bL, (short)0, cacc[nt], false, false);
            cacc[nt] = __builtin_amdgcn_wmma_f32_16x16x4_f32(
                false, aH, false, bR, (short)0, cacc[nt], false, false);
        }
    }
#pragma unroll
    for (int nt = 0; nt < 4; ++nt) {
        int col = nt * 16 + m;
        float b = bl[col];
#pragma unroll
        for (int i = 0; i < 8; ++i) {             // C/D layout: vgpr i = rows i, i+8
            hout[(size_t)(row0 + i + 8 * kh) * 64 + col] = cacc[nt][i] + b;
        }
    }
}

// ----------------------------------------------------------- segment-max pool
__device__ __forceinline__ unsigned f2ord(float f) {
    unsigned b = __float_as_uint(f);
    return (b & 0x80000000u) ? ~b : (b | 0x80000000u);
}
__device__ __forceinline__ float ord2f(unsigned o) {
    unsigned b = (o & 0x80000000u) ? (o ^ 0x80000000u) : ~o;
    return __uint_as_float(b);
}

__global__ void k_pool_init(unsigned* __restrict__ pooled) {
    int i = blockIdx.x * blockDim.x + threadIdx.x;
    if (i < NG * 64) pooled[i] = 0x007FFFFFu;     // ord(-inf)
}

__global__ void k_pool(const float* __restrict__ h, const int* __restrict__ batch,
                       unsigned* __restrict__ pooled) {
    int idx = blockIdx.x * blockDim.x + threadIdx.x;
    if (idx < NN * 64) {
        int n = idx >> 6, f = idx & 63;
        atomicMax(&pooled[batch[n] * 64 + f], f2ord(h[idx]));
    }
}

// ---------------------------------------------------- head: GEMM + log_softmax
__global__ void k_head(const unsigned* __restrict__ pooled, const float* __restrict__ W,
                       const float* __restrict__ b, float* __restrict__ out) {
    int g = threadIdx.x;
    if (g >= NG) return;
    float p[64];
#pragma unroll
    for (int j = 0; j < 64; ++j) p[j] = ord2f(pooled[g * 64 + j]);
    float logits[8];
#pragma unroll
    for (int k = 0; k < 8; ++k) {
        float a = b[k];
        for (int j = 0; j < 64; ++j) a += p[j] * W[j * 8 + k];
        logits[k] = a;
    }
    float mx = logits[0];
#pragma unroll
    for (int k = 1; k < 8; ++k) mx = fmaxf(mx, logits[k]);
    float s = 0.0f;
#pragma unroll
    for (int k = 0; k < 8; ++k) s += __expf(logits[k] - mx);
    float lse = __logf(s);
#pragma unroll
    for (int k = 0; k < 8; ++k) out[g * 8 + k] = logits[k] - mx - lse;
}

// ---------------------------------------------------------------------- driver
extern "C" void kernel_launch(void* const* d_in, const int* in_sizes, int n_in,
                              void* d_out, int out_size, void* d_ws, size_t ws_size,
                              hipStream_t stream) {
    const float* x     = (const float*)d_in[0];
    const float* Wl0   = (const float*)d_in[1];
    const float* bl0   = (const float*)d_in[2];
    const float* Wr0   = (const float*)d_in[3];
    const float* Wl1   = (const float*)d_in[4];
    const float* bl1   = (const float*)d_in[5];
    const float* Wr1   = (const float*)d_in[6];
    const float* Wl2   = (const float*)d_in[7];
    const float* bl2   = (const float*)d_in[8];
    const float* Wr2   = (const float*)d_in[9];
    const float* W_out = (const float*)d_in[10];
    const float* b_out = (const float*)d_in[11];
    const int*   ei    = (const int*)d_in[12];
    const int*   batch = (const int*)d_in[13];
    const int* src = ei;
    const int* dst = ei + NE;

    // workspace partition (256B aligned)
    char* ws = (char*)d_ws;
    size_t off = 0;
    auto take = [&](size_t bytes) { char* p = ws + off; off = (off + bytes + 255) & ~(size_t)255; return p; };
    float*    cnt    = (float*)take((size_t)NN * 4);        // degree -> inv degree
    float*    accb   = (float*)take((size_t)NN * 64 * 4);   // aggregation sums
    float*    hA     = (float*)take((size_t)NN * 64 * 4);
    float*    hB     = (float*)take((size_t)NN * 64 * 4);
    unsigned* pooled = (unsigned*)take((size_t)NG * 64 * 4);

    // degree + reciprocal (edge structure fixed for all 3 layers)
    hipMemsetAsync(cnt, 0, (size_t)NN * 4, stream);
    k_count<<<(NE + 255) / 256, 256, 0, stream>>>(dst, cnt);
    k_inv<<<(NN + 255) / 256, 256, 0, stream>>>(cnt);

    const int AGG_BLOCKS  = 8192;                            // 65536 waves grid-stride
    const int GEMM_BLOCKS = (NN / 16 + 7) / 8;               // 8 waves/block

    // layer 0: F=32 -> 64
    hipMemsetAsync(accb, 0, (size_t)NN * 32 * 4, stream);
    k_aggregate<32><<<AGG_BLOCKS, 256, 0, stream>>>(x, src, dst, accb);
    k_sage_gemm<32><<<GEMM_BLOCKS, 256, 0, stream>>>(accb, cnt, x, Wl0, bl0, Wr0, hA);

    // layer 1: 64 -> 64
    hipMemsetAsync(accb, 0, (size_t)NN * 64 * 4, stream);
    k_aggregate<64><<<AGG_BLOCKS, 256, 0, stream>>>(hA, src, dst, accb);
    k_sage_gemm<64><<<GEMM_BLOCKS, 256, 0, stream>>>(accb, cnt, hA, Wl1, bl1, Wr1, hB);

    // layer 2: 64 -> 64
    hipMemsetAsync(accb, 0, (size_t)NN * 64 * 4, stream);
    k_aggregate<64><<<AGG_BLOCKS, 256, 0, stream>>>(hB, src, dst, accb);
    k_sage_gemm<64><<<GEMM_BLOCKS, 256, 0, stream>>>(accb, cnt, hB, Wl2, bl2, Wr2, hA);

    // pool + head
    k_pool_init<<<(NG * 64 + 255) / 256, 256, 0, stream>>>(pooled);
    k_pool<<<(NN * 64 + 255) / 256, 256, 0, stream>>>(hA, batch, pooled);
    k_head<<<1, 64, 0, stream>>>(pooled, W_out, b_out, (float*)d_out);
}